// _FastMultiTaskGP_80315888435598
// MI455X (gfx1250) — compile-verified
//
#include <hip/hip_runtime.h>
#include <math.h>

// ---------------------------------------------------------------------------
// FastMultiTaskGP pipeline for MI455X (gfx1250, wave32):
//   K1: stage-1 of four-step FFT (256-pt FFTs over n2) via V_WMMA_F32_16X16X4_F32
//   K2: stage-2 (256-pt FFTs over n1) via WMMA, produces lams / yt
//   K3: per-frequency Schur bordering recursion (literal reference math),
//       exploiting conjugate symmetry: A(N-n) = conj(A(n))
//   K4: deterministic final reduction of logdet / quad
// Twiddles come from per-block LDS root-of-unity tables (no per-element sincos).
// ---------------------------------------------------------------------------

typedef __attribute__((ext_vector_type(2))) float v2f;
typedef __attribute__((ext_vector_type(8))) float v8f;

#define N_FFT    65536
#define NSEQ_TRI 136            // upper-triangle (i<=j) sequences of k1
#define NSEQ     152            // + 16 rows of y
#define M2       (16*16*65536)  // one output plane (A.real or A.imag)
#define NHALF    32769          // n in [0, 32768] computed, rest mirrored

// ---------------- complex helpers (float2) ----------------
__device__ __forceinline__ float2 cmul(float2 a, float2 b) {
    return make_float2(a.x*b.x - a.y*b.y, a.x*b.y + a.y*b.x);
}
__device__ __forceinline__ float2 cmulc(float2 a, float2 b) { // a * conj(b)
    return make_float2(a.x*b.x + a.y*b.y, a.y*b.x - a.x*b.y);
}
__device__ __forceinline__ float2 cadd(float2 a, float2 b) { return make_float2(a.x+b.x, a.y+b.y); }
__device__ __forceinline__ float2 csub(float2 a, float2 b) { return make_float2(a.x-b.x, a.y-b.y); }
__device__ __forceinline__ float  cabs2(float2 a)          { return a.x*a.x + a.y*a.y; }
__device__ __forceinline__ float2 cinv(float2 a) {
    float d = 1.0f / (a.x*a.x + a.y*a.y);
    return make_float2(a.x*d, -a.y*d);
}

// D(16x16) += A(16x4) * B(4x16), f32 in/out (CDNA5 WMMA)
__device__ __forceinline__ v8f wmma4(v2f a, v2f b, v8f c) {
    return __builtin_amdgcn_wmma_f32_16x16x4_f32(false, a, false, b, (short)0, c, false, false);
}

// sequence id -> (i,j) with i<=j, t = j*(j+1)/2 + i
__device__ __forceinline__ void seq_to_ij(int s, int& i, int& j) {
    int jj = 0;
    while ((jj + 1) * (jj + 2) / 2 <= s) jj++;
    j = jj;
    i = s - jj * (jj + 1) / 2;
}

// ---------------------------------------------------------------------------
// Stage 1: for each (seq, n1): 256-pt FFT over n2 of x[n1 + 256*n2] (real in),
// then multiply by W_N^{n1*k2}; store w[seq][k2][n1].
// One wave32 per FFT.  256-FFT = F16 * (W256 o (D * F16)) with 16x16 tiles.
// WMMA layouts (ISA 7.12.2):
//   A 16x4 : lane L, reg r -> A[L&15][r + 2*(L>>4)]
//   B 4x16 : lane L, reg r -> B[r + 2*(L>>4)][L&15]
//   C 16x16: lane L, reg r -> C[r + 8*(L>>4)][L&15]
// ---------------------------------------------------------------------------
__global__ __launch_bounds__(256) void fft_stage1(const float* __restrict__ k1,
                                                  const float* __restrict__ y,
                                                  float2* __restrict__ w)
{
    __shared__ float2 tile[8][16][17];   // per-wave padded 16x16 complex tile
    __shared__ float2 w16t[16];          // exp(-2*pi*i * t/16)
    __shared__ float2 coarse[256];       // exp(-2*pi*i * t/256)
    __shared__ float2 fine[256];         // exp(-2*pi*i * t/65536)

    const int tid = threadIdx.x;
    {   // build root-of-unity tables once per block (one sincospi per table entry)
        float s, c;
        sincospif((float)tid * (1.0f / 128.0f), &s, &c);
        coarse[tid] = make_float2(c, -s);
        sincospif((float)tid * (1.0f / 32768.0f), &s, &c);
        fine[tid] = make_float2(c, -s);
        if (tid < 16) {
            sincospif((float)tid * 0.125f, &s, &c);
            w16t[tid] = make_float2(c, -s);
        }
    }
    __syncthreads();

    const int lane = tid & 31;
    const int wv   = tid >> 5;
    const int wid  = blockIdx.x * 8 + wv;
    const int seq  = wid >> 8;
    const int n1   = wid & 255;
    const int m    = lane & 15;
    const int h    = lane >> 4;

    const float* base;
    if (seq < NSEQ_TRI) {
        int i, j; seq_to_ij(seq, i, j);
        base = k1 + (size_t)(i * 16 + j) * N_FFT;
    } else {
        base = y + (size_t)(seq - NSEQ_TRI) * N_FFT;
    }

    // E = D * F16   (D real -> 8 WMMAs)
    v8f er = {0,0,0,0,0,0,0,0}, ei = {0,0,0,0,0,0,0,0};
#pragma unroll
    for (int kk = 0; kk < 4; ++kk) {
        v2f dr, bc, bns;
#pragma unroll
        for (int r = 0; r < 2; ++r) {
            int k = kk*4 + r + 2*h;                       // A col / B row
            dr[r] = base[(size_t)n1 + 256u * (unsigned)(m + 16*k)];
            float2 f = w16t[(k * m) & 15];                // (cos, -sin)
            bc[r]  = f.x;                                 // Fr
            bns[r] = f.y;                                 // Fi = -sin
        }
        er = wmma4(dr, bc,  er);
        ei = wmma4(dr, bns, ei);
    }

    // twiddle W256^{a*c}, stash to LDS (C/D layout -> tile[row][col])
#pragma unroll
    for (int r = 0; r < 8; ++r) {
        int a = r + 8*h, c = m;
        tile[wv][a][c] = cmul(make_float2(er[r], ei[r]), coarse[a * c]);
    }
    __syncthreads();

    // V = F16 * E'   (16 WMMAs)
    v8f vr = {0,0,0,0,0,0,0,0}, vi = {0,0,0,0,0,0,0,0};
#pragma unroll
    for (int kk = 0; kk < 4; ++kk) {
        v2f fc, fs, fns, ber, bei;
#pragma unroll
        for (int r = 0; r < 2; ++r) {
            int k = kk*4 + r + 2*h;
            float2 f = w16t[(m * k) & 15];                // F[m][k] = (cos, -sin)
            fc[r] = f.x; fs[r] = -f.y; fns[r] = f.y;
            float2 e = tile[wv][k][m];                    // B frag of E'
            ber[r] = e.x; bei[r] = e.y;
        }
        vr = wmma4(fc,  ber, vr);   // Fr*Er
        vr = wmma4(fs,  bei, vr);   // -Fi*Ei = +sin*Ei
        vi = wmma4(fc,  bei, vi);   // Fr*Ei
        vi = wmma4(fns, ber, vi);   // Fi*Er = -sin*Er
    }
    __syncthreads();

    // big twiddle W_N^{n1*k2} = coarse[x>>8]*fine[x&255], transposed store
#pragma unroll
    for (int r = 0; r < 8; ++r) {
        int d = r + 8*h, c = m;
        int k2 = c + 16*d;
        int x  = n1 * k2;                                 // < 65536
        float2 tw = cmul(coarse[x >> 8], fine[x & 255]);
        float2 v = cmul(make_float2(vr[r], vi[r]), tw);
        w[((size_t)seq * 256 + k2) * 256 + n1] = v;
    }
}

// ---------------------------------------------------------------------------
// Stage 2: for each (seq, k2): 256-pt FFT over n1 of w[seq][k2][n1] (complex),
// write lams[seq][k2 + 256*k1] (ortho 1/256 scale for y rows).
// ---------------------------------------------------------------------------
__global__ __launch_bounds__(256) void fft_stage2(const float2* __restrict__ w,
                                                  float2* __restrict__ lams)
{
    __shared__ float2 tile[8][16][17];
    __shared__ float2 w16t[16];
    __shared__ float2 coarse[256];

    const int tid = threadIdx.x;
    {
        float s, c;
        sincospif((float)tid * (1.0f / 128.0f), &s, &c);
        coarse[tid] = make_float2(c, -s);
        if (tid < 16) {
            sincospif((float)tid * 0.125f, &s, &c);
            w16t[tid] = make_float2(c, -s);
        }
    }
    __syncthreads();

    const int lane = tid & 31;
    const int wv   = tid >> 5;
    const int wid  = blockIdx.x * 8 + wv;
    const int seq  = wid >> 8;
    const int k2   = wid & 255;
    const int m    = lane & 15;
    const int h    = lane >> 4;

    const float2* v = w + ((size_t)seq * 256 + k2) * 256;

    // E = D * F16, complex D (16 WMMAs)
    v8f er = {0,0,0,0,0,0,0,0}, ei = {0,0,0,0,0,0,0,0};
#pragma unroll
    for (int kk = 0; kk < 4; ++kk) {
        v2f drr, dii, bc, bs, bns;
#pragma unroll
        for (int r = 0; r < 2; ++r) {
            int k = kk*4 + r + 2*h;
            float2 d = v[m + 16*k];                        // coalesced per half-wave
            drr[r] = d.x; dii[r] = d.y;
            float2 f = w16t[(k * m) & 15];
            bc[r] = f.x; bs[r] = -f.y; bns[r] = f.y;
        }
        er = wmma4(drr, bc,  er);   // Dr*cos
        er = wmma4(dii, bs,  er);   // +Di*sin   (= -Di*Fi)
        ei = wmma4(dii, bc,  ei);   // Di*cos
        ei = wmma4(drr, bns, ei);   // -Dr*sin   (= Dr*Fi)
    }

#pragma unroll
    for (int r = 0; r < 8; ++r) {
        int a = r + 8*h, c = m;
        tile[wv][a][c] = cmul(make_float2(er[r], ei[r]), coarse[a * c]);
    }
    __syncthreads();

    v8f vr = {0,0,0,0,0,0,0,0}, vi = {0,0,0,0,0,0,0,0};
#pragma unroll
    for (int kk = 0; kk < 4; ++kk) {
        v2f fc, fs, fns, ber, bei;
#pragma unroll
        for (int r = 0; r < 2; ++r) {
            int k = kk*4 + r + 2*h;
            float2 f = w16t[(m * k) & 15];
            fc[r] = f.x; fs[r] = -f.y; fns[r] = f.y;
            float2 e = tile[wv][k][m];
            ber[r] = e.x; bei[r] = e.y;
        }
        vr = wmma4(fc,  ber, vr);
        vr = wmma4(fs,  bei, vr);
        vi = wmma4(fc,  bei, vi);
        vi = wmma4(fns, ber, vi);
    }
    __syncthreads();

    const float scale = (seq >= NSEQ_TRI) ? (1.0f / 256.0f) : 1.0f;  // ortho for y
#pragma unroll
    for (int r = 0; r < 8; ++r) {
        int d = r + 8*h, c = m;
        int k1i = c + 16*d;
        int n = k2 + 256 * k1i;
        lams[(size_t)seq * N_FFT + n] = make_float2(vr[r] * scale, vi[r] * scale);
    }
}

// ---------------------------------------------------------------------------
// K3: per-frequency Schur bordering recursion (verbatim reference math).
// One thread per frequency n in [0, 32768]; A(N-n) = conj(A(n)) mirrored out.
// Per-thread 16x16 complex A in LDS (stride 289 float2 -> conflict-free).
// ---------------------------------------------------------------------------
__global__ __launch_bounds__(32) void schur_kernel(const float2* __restrict__ lams,
                                                   float* __restrict__ out,
                                                   float2* __restrict__ partials)
{
    extern __shared__ float2 sm[];
    const int tid = threadIdx.x;                 // 0..31
    const int n   = blockIdx.x * 32 + tid;       // frequency
    const bool active = (n < NHALF);

    float logdet = 0.f, quad = 0.f;

    if (active) {
        float2* A  = sm + (size_t)tid * 289;     // 256 A + 16 B + 16 Tm + pad
        float2* Bv = A + 256;
        float2* Tm = Bv + 16;

        float2 l00 = lams[n];                    // tri(0,0) = 0
        logdet = 0.5f * logf(cabs2(l00));
        A[0] = cinv(l00);

        for (int l = 1; l < 16; ++l) {
            const int tbase = l * (l + 1) / 2;
            for (int i = 0; i < l; ++i)
                Bv[i] = lams[(size_t)(tbase + i) * N_FFT + n];

            for (int i = 0; i < l; ++i) {        // Tm = A * B
                float2 acc = make_float2(0.f, 0.f);
                for (int k = 0; k < l; ++k) acc = cadd(acc, cmul(A[i*16 + k], Bv[k]));
                Tm[i] = acc;
            }
            float2 M = make_float2(0.f, 0.f);    // M = sum conj(B)*Tm
            for (int i = 0; i < l; ++i) M = cadd(M, cmulc(Tm[i], Bv[i]));

            float2 S = csub(lams[(size_t)(tbase + l) * N_FFT + n], M);
            logdet += 0.5f * logf(cabs2(S));
            float2 Sinv = cinv(S);

            for (int i = 0; i < l; ++i) {
                float2 P = cmul(Tm[i], Sinv);
                for (int k = 0; k < l; ++k)      // C = A + P * conj(Tm)^T
                    A[i*16 + k] = cadd(A[i*16 + k], cmulc(P, Tm[k]));
                A[i*16 + l] = make_float2(-P.x, -P.y);      // -P
            }
            for (int k = 0; k < l; ++k) {
                float2 P = cmul(Tm[k], Sinv);
                A[l*16 + k] = make_float2(-P.x, P.y);       // -conj(P)
            }
            A[l*16 + l] = Sinv;
        }

        // quad = real( conj(yt_i) * A_ij * yt_j )
        float2 yl[16];
#pragma unroll
        for (int i = 0; i < 16; ++i)
            yl[i] = lams[(size_t)(NSEQ_TRI + i) * N_FFT + n];

        for (int i = 0; i < 16; ++i) {
            float2 acc = make_float2(0.f, 0.f);
#pragma unroll
            for (int j = 0; j < 16; ++j) acc = cadd(acc, cmul(A[i*16 + j], yl[j]));
            quad += yl[i].x * acc.x + yl[i].y * acc.y;
        }

        // mirror weight: interior frequencies stand for themselves and N-n
        const bool mir = (n != 0) && (n != N_FFT / 2);
        const float wgt = mir ? 2.f : 1.f;
        logdet *= wgt;
        quad   *= wgt;

        // write A planes: out[(i*16+j)*N + n] (real), + M2 (imag);
        // mirrored plane gets conj(A)
        for (int i = 0; i < 16; ++i)
            for (int j = 0; j < 16; ++j) {
                float2 a = A[i*16 + j];
                size_t pb = (size_t)(i * 16 + j) * N_FFT;
                out[pb + n]                 = a.x;
                out[(size_t)M2 + pb + n]    = a.y;
                if (mir) {
                    size_t nm = (size_t)(N_FFT - n);
                    out[pb + nm]              = a.x;
                    out[(size_t)M2 + pb + nm] = -a.y;
                }
            }
    }

    // wave32 reduction of (logdet, quad) -- all lanes participate
#pragma unroll
    for (int off = 16; off > 0; off >>= 1) {
        logdet += __shfl_down(logdet, off, 32);
        quad   += __shfl_down(quad,   off, 32);
    }
    if (tid == 0) partials[blockIdx.x] = make_float2(logdet, quad);
}

// K4: deterministic serial reduction
__global__ void reduce_final(const float2* __restrict__ partials,
                             float* __restrict__ out, int nblocks)
{
    if (threadIdx.x == 0 && blockIdx.x == 0) {
        double ld = 0.0, q = 0.0;
        for (int i = 0; i < nblocks; ++i) { ld += partials[i].x; q += partials[i].y; }
        out[(size_t)2 * M2]     = (float)ld;
        out[(size_t)2 * M2 + 1] = (float)q;
    }
}

// ---------------------------------------------------------------------------
extern "C" void kernel_launch(void* const* d_in, const int* in_sizes, int n_in,
                              void* d_out, int out_size, void* d_ws, size_t ws_size,
                              hipStream_t stream)
{
    (void)in_sizes; (void)n_in; (void)out_size; (void)ws_size;

    const float* k1 = (const float*)d_in[0];
    const float* y  = (const float*)d_in[1];
    float*  out      = (float*)d_out;
    float2* w        = (float2*)d_out;                       // stage-1 scratch: 9.96M float2 fits in 16.7M float2
    float2* lams     = (float2*)d_ws;                        // 152*65536 float2 ~ 79.7 MB
    float2* partials = lams + (size_t)NSEQ * N_FFT;          // per-block partial sums

    const int totalWaves = NSEQ * 256;                       // 38912
    dim3 blk(256);
    dim3 gfft(totalWaves / 8);                               // 8 waves per block

    fft_stage1<<<gfft, blk, 0, stream>>>(k1, y, w);
    fft_stage2<<<gfft, blk, 0, stream>>>(w, lams);

    const int nblocks = (NHALF + 31) / 32;                   // 1025 (covers n in [0,32768])
    size_t shmem = (size_t)32 * 289 * sizeof(float2);        // 73,984 B (<= 320 KB WGP LDS)
    schur_kernel<<<dim3(nblocks), dim3(32), shmem, stream>>>(lams, out, partials);

    reduce_final<<<dim3(1), dim3(64), 0, stream>>>(partials, out, nblocks);
}